// EEGGCN_44624710205899
// MI455X (gfx1250) — compile-verified
//
#include <hip/hip_runtime.h>
#include <hip/hip_bf16.h>
#include <stdint.h>

// ---------------- problem constants (from reference) ----------------
#define NN     16384       // nodes
#define GG     256         // graphs
#define EE     262144      // directed edges (before self loops)
#define FIN    3000
#define FH     640
#define FOUT   64
#define NPG    (NN / GG)   // 64 nodes per graph
#define BN_SCALE 0.99999500003750f   // 1/sqrt(1 + 1e-5)
#define LEAKY    0.01f

typedef float v2f __attribute__((ext_vector_type(2)));
typedef float v8f __attribute__((ext_vector_type(8)));

__device__ __forceinline__ uint32_t lds_offset(const void* p) {
    // generic LDS pointer = {SHARED_BASE aperture, 32-bit LDS offset}
    return (uint32_t)(uintptr_t)p;
}

__device__ __forceinline__ void async_copy_b128(const void* gptr, uint32_t lds) {
    asm volatile("global_load_async_to_lds_b128 %0, %1, off"
                 :: "v"(lds), "v"(gptr) : "memory");
}

__device__ __forceinline__ void wait_async0() {
    asm volatile("s_wait_asynccnt 0x0" ::: "memory");
}

// ---------------- utility: zero fill ----------------
__global__ __launch_bounds__(256) void zero_f32(float* __restrict__ p, size_t n) {
    size_t i = (size_t)blockIdx.x * blockDim.x + threadIdx.x;
    size_t stride = (size_t)gridDim.x * blockDim.x;
    for (; i < n; i += stride) p[i] = 0.0f;
}

// ---------------- degree count (dst side; self-loop added later) ----------------
__global__ __launch_bounds__(256) void deg_kernel(const int* __restrict__ dst,
                                                  float* __restrict__ deg) {
    int e = blockIdx.x * blockDim.x + threadIdx.x;
    if (e < EE) unsafeAtomicAdd(&deg[dst[e]], 1.0f);
}

__global__ __launch_bounds__(256) void dinv_kernel(const float* __restrict__ deg,
                                                   float* __restrict__ dinv) {
    int n = blockIdx.x * blockDim.x + threadIdx.x;
    if (n < NN) dinv[n] = rsqrtf(deg[n] + 1.0f);   // +1 self loop
}

// ---------------- f32 WMMA GEMM: C[M,Ncols] = A[M,K] * B[K,Ncols] ----------------
// Block tile 128x64, K-step 16, double-buffered LDS filled by async global->LDS DMA
// (GLOBAL_LOAD_ASYNC_TO_LDS_B128, ASYNCcnt). 8 waves, each wave owns a 32x32 patch
// (2x2 tiles of v_wmma_f32_16x16x4_f32, exact f32 accumulation).
#define BM 128
#define BNT 64
#define BK 16

__global__ __launch_bounds__(256) void gemm_wmma_f32(
    const float* __restrict__ A, const float* __restrict__ B,
    float* __restrict__ C, int M, int K, int Ncols)
{
    __shared__ float sA[2][BM][BK + 1];    // +1 pad: stride 17 dwords
    __shared__ float sB[2][BK][BNT + 4];   // +4 pad

    const int tid  = threadIdx.x;
    const int lane = tid & 31;
    const int wid  = tid >> 5;          // 0..7
    const int wm   = wid & 3;           // wave row: m offset wm*32
    const int wn   = wid >> 2;          // wave col: n offset wn*32
    const int blockRow = blockIdx.x * BM;
    const int blockCol = blockIdx.y * BNT;

    const int lhalf = lane & 15;
    const int khi   = (lane >> 4) << 1;  // 0 for lanes 0-15, 2 for lanes 16-31

    // per-thread tile-load coordinates (fixed across K iterations)
    const int ar0 = tid >> 2;            // A rows handled: ar0, ar0+64
    const int ac4 = (tid & 3) << 2;      // A col within chunk: 0,4,8,12
    const int br  = tid >> 4;            // B row within chunk: 0..15
    const int bc4 = (tid & 15) << 2;     // B col: 0..60

    // precomputed global base pointers and LDS destinations (both buffers)
    const float* gA0 = A + (size_t)(blockRow + ar0) * K + ac4;       // + kb
    const float* gA1 = A + (size_t)(blockRow + ar0 + 64) * K + ac4;  // + kb
    const float* gB  = B + (size_t)br * Ncols + blockCol + bc4;      // + kb*Ncols
    const uint32_t ldsA0[2] = { lds_offset(&sA[0][ar0][ac4]),      lds_offset(&sA[1][ar0][ac4]) };
    const uint32_t ldsA1[2] = { lds_offset(&sA[0][ar0 + 64][ac4]), lds_offset(&sA[1][ar0 + 64][ac4]) };
    const uint32_t ldsB [2] = { lds_offset(&sB[0][br][bc4]),       lds_offset(&sB[1][br][bc4]) };

    // full chunk: unconditional async DMA, no bounds checks
    auto load_full = [&](int kb, int buf) {
        async_copy_b128(gA0 + kb, ldsA0[buf]);
        async_copy_b128(gA1 + kb, ldsA1[buf]);
        async_copy_b128(gB + (size_t)kb * Ncols, ldsB[buf]);
    };
    // tail chunk (runs at most once per kernel): guarded synchronous loads
    auto load_tail = [&](int kb, int buf) {
        #pragma unroll
        for (int it = 0; it < 2; ++it) {
            int r = ar0 + it * 64;
            const float* gp = (it == 0 ? gA0 : gA1) + kb;
            #pragma unroll
            for (int j = 0; j < 4; ++j)
                sA[buf][r][ac4 + j] = (kb + ac4 + j < K) ? gp[j] : 0.0f;
        }
        const float* bp = gB + (size_t)kb * Ncols;
        bool inb = (kb + br < K);
        #pragma unroll
        for (int j = 0; j < 4; ++j)
            sB[buf][br][bc4 + j] = inb ? bp[j] : 0.0f;
    };
    auto load_chunk = [&](int kb, int buf) {   // uniform scalar branch
        if (kb + BK <= K) load_full(kb, buf);
        else              load_tail(kb, buf);
    };

    const int nChunks = (K + BK - 1) / BK;

    v8f acc[2][2] = {};

    int buf = 0;
    load_chunk(0, 0);
    wait_async0();
    __syncthreads();

    for (int c = 0; c < nChunks; ++c) {
        const bool more = (c + 1 < nChunks);
        if (more) load_chunk((c + 1) * BK, buf ^ 1);   // prefetch into other buffer

        const int mb = wm * 32;
        const int nb = wn * 32;
        #pragma unroll
        for (int kk = 0; kk < BK; kk += 4) {
            // A frag (16x4 f32): lanes 0-15 hold K=kk,kk+1; lanes 16-31 hold K=kk+2,kk+3
            v2f a[2], b[2];
            #pragma unroll
            for (int i = 0; i < 2; ++i) {
                const float* ap = &sA[buf][mb + i * 16 + lhalf][kk + khi];
                a[i].x = ap[0]; a[i].y = ap[1];
            }
            // B frag (4x16 f32): vgpr0 = rows kk / kk+2, vgpr1 = rows kk+1 / kk+3
            #pragma unroll
            for (int j = 0; j < 2; ++j) {
                b[j].x = sB[buf][kk + khi + 0][nb + j * 16 + lhalf];
                b[j].y = sB[buf][kk + khi + 1][nb + j * 16 + lhalf];
            }
            #pragma unroll
            for (int i = 0; i < 2; ++i)
                #pragma unroll
                for (int j = 0; j < 2; ++j)
                    acc[i][j] = __builtin_amdgcn_wmma_f32_16x16x4_f32(
                        false, a[i], false, b[j], (short)0, acc[i][j], false, false);
        }

        if (more) wait_async0();              // next chunk's DMA complete
        __syncthreads();                      // everyone done reading `buf`
        buf ^= 1;
    }

    // ---- store: C/D layout: vgpr v -> row v (lanes 0-15) / row v+8 (lanes 16-31) ----
    const int radd = (lane >> 4) * 8;
    #pragma unroll
    for (int i = 0; i < 2; ++i) {
        #pragma unroll
        for (int j = 0; j < 2; ++j) {
            int row0 = blockRow + wm * 32 + i * 16 + radd;
            int col  = blockCol + wn * 32 + j * 16 + lhalf;
            #pragma unroll
            for (int v = 0; v < 8; ++v)
                C[(size_t)(row0 + v) * Ncols + col] = acc[i][j][v];
        }
    }
}

// ---------------- edge aggregation: agg[dst] += hw[src] * dinv[src]*dinv[dst] ----------------
// One wave per edge; float4 coalesced gather, hardware f32 atomics (L2-resident: 42MB << 192MB L2).
__global__ __launch_bounds__(256) void edge_agg(
    const int* __restrict__ src, const int* __restrict__ dst,
    const float* __restrict__ dinv, const float* __restrict__ hw,
    float* __restrict__ agg, int F)
{
    int e = blockIdx.x * 8 + (threadIdx.x >> 5);
    if (e >= EE) return;
    int lane = threadIdx.x & 31;
    int s = src[e], d = dst[e];
    float w = dinv[s] * dinv[d];
    const float4* hp = (const float4*)(hw + (size_t)s * F);
    float* ap = agg + (size_t)d * F;
    int nf4 = F >> 2;
    for (int i = lane; i < nf4; i += 32) {
        float4 v = hp[i];
        int base = i << 2;
        unsafeAtomicAdd(ap + base + 0, v.x * w);
        unsafeAtomicAdd(ap + base + 1, v.y * w);
        unsafeAtomicAdd(ap + base + 2, v.z * w);
        unsafeAtomicAdd(ap + base + 3, v.w * w);
    }
}

// ---------------- fused epilogue: self-loop + bias + BN(eval) + leaky-relu ----------------
// h may alias agg (pure elementwise, read-once).
__global__ __launch_bounds__(256) void epilogue(
    const float* __restrict__ hw, const float* agg, const float* __restrict__ dinv,
    const float* __restrict__ bias, const float* __restrict__ gamma,
    const float* __restrict__ beta, float* h, int F)
{
    int n = blockIdx.x;
    float di = dinv[n];
    float di2 = di * di;
    for (int f = threadIdx.x; f < F; f += blockDim.x) {
        size_t idx = (size_t)n * F + f;
        float v = agg[idx] + hw[idx] * di2 + bias[f];
        v = v * (gamma[f] * BN_SCALE) + beta[f];
        h[idx] = v > 0.0f ? v : v * LEAKY;
    }
}

// ---------------- global max pool per graph + linear head ----------------
__global__ __launch_bounds__(64) void pool_head(
    const float* __restrict__ h, const float* __restrict__ Wm,
    const float* __restrict__ bm, float* __restrict__ out)
{
    __shared__ float red[64];
    int g = blockIdx.x;
    int f = threadIdx.x;                       // FOUT == 64
    const float* base = h + (size_t)g * NPG * FOUT + f;
    float m = -INFINITY;
    #pragma unroll
    for (int i = 0; i < NPG; ++i) m = fmaxf(m, base[(size_t)i * FOUT]);
    red[f] = m * Wm[f];
    __syncthreads();
    for (int s = 32; s > 0; s >>= 1) {
        if (f < s) red[f] += red[f + s];
        __syncthreads();
    }
    if (f == 0) out[g] = red[0] + bm[0];
}

// ---------------- launch ----------------
extern "C" void kernel_launch(void* const* d_in, const int* in_sizes, int n_in,
                              void* d_out, int out_size, void* d_ws, size_t ws_size,
                              hipStream_t stream) {
    const float* x   = (const float*)d_in[0];
    const int*   ei  = (const int*)d_in[1];    // [2,E]: row 0 = src, row 1 = dst
    // d_in[2] = batch (contiguous 64-node groups; used implicitly)
    const float* W0  = (const float*)d_in[3];
    const float* b0  = (const float*)d_in[4];
    const float* g0  = (const float*)d_in[5];
    const float* be0 = (const float*)d_in[6];
    const float* W1  = (const float*)d_in[7];
    const float* b1  = (const float*)d_in[8];
    const float* g1  = (const float*)d_in[9];
    const float* be1 = (const float*)d_in[10];
    const float* W2  = (const float*)d_in[11];
    const float* b2  = (const float*)d_in[12];
    const float* g2  = (const float*)d_in[13];
    const float* be2 = (const float*)d_in[14];
    const float* Wm  = (const float*)d_in[15];
    const float* bm  = (const float*)d_in[16];
    float* out = (float*)d_out;

    float* ws   = (float*)d_ws;
    float* bufA = ws;                                   // N*H  (hw0 / hw1)
    float* bufB = bufA + (size_t)NN * FH;               // N*H  (agg / h, in place)
    float* bufC = bufB + (size_t)NN * FH;               // N*OUT (hw2)
    float* bufD = bufC + (size_t)NN * FOUT;             // N*OUT (agg2 / h2)
    float* degb = bufD + (size_t)NN * FOUT;             // N
    float* dinv = degb + NN;                            // N

    const int* esrc = ei;
    const int* edst = ei + EE;

    // ---- symmetric normalization ----
    zero_f32<<<64, 256, 0, stream>>>(degb, (size_t)NN);
    deg_kernel<<<EE / 256, 256, 0, stream>>>(edst, degb);
    dinv_kernel<<<NN / 256, 256, 0, stream>>>(degb, dinv);

    dim3 gemmGridH(NN / BM, FH / BNT);    // (128, 10)
    dim3 gemmGridO(NN / BM, FOUT / BNT);  // (128, 1)

    // ---- layer 0 ----
    gemm_wmma_f32<<<gemmGridH, 256, 0, stream>>>(x, W0, bufA, NN, FIN, FH);
    zero_f32<<<4096, 256, 0, stream>>>(bufB, (size_t)NN * FH);
    edge_agg<<<EE / 8, 256, 0, stream>>>(esrc, edst, dinv, bufA, bufB, FH);
    epilogue<<<NN, 256, 0, stream>>>(bufA, bufB, dinv, b0, g0, be0, bufB, FH);

    // ---- layer 1 ----
    gemm_wmma_f32<<<gemmGridH, 256, 0, stream>>>(bufB, W1, bufA, NN, FH, FH);
    zero_f32<<<4096, 256, 0, stream>>>(bufB, (size_t)NN * FH);
    edge_agg<<<EE / 8, 256, 0, stream>>>(esrc, edst, dinv, bufA, bufB, FH);
    epilogue<<<NN, 256, 0, stream>>>(bufA, bufB, dinv, b1, g1, be1, bufB, FH);

    // ---- layer 2 ----
    gemm_wmma_f32<<<gemmGridO, 256, 0, stream>>>(bufB, W2, bufC, NN, FH, FOUT);
    zero_f32<<<1024, 256, 0, stream>>>(bufD, (size_t)NN * FOUT);
    edge_agg<<<EE / 8, 256, 0, stream>>>(esrc, edst, dinv, bufC, bufD, FOUT);
    epilogue<<<NN, 256, 0, stream>>>(bufC, bufD, dinv, b2, g2, be2, bufD, FOUT);

    // ---- pool + head ----
    pool_head<<<GG, 64, 0, stream>>>(bufD, Wm, bm, out);
}